// FactorizedTargetPositionPredictor_19361712570425
// MI455X (gfx1250) — compile-verified
//
#include <hip/hip_runtime.h>
#include <math.h>

// ---- problem constants ----
#define MULK   128
#define NCH    4
#define NRAD   64
#define RESB   180
#define RESA   359
#define NG     16
#define SHD    36
#define EMBD   4608
#define NBA    (RESB * RESA)      // 64620 grid points (divisible by 4)
#define NB_N   505                // 128-column blocks per M-group
#define NPAD2  (NB_N * 128)       // 64640 padded columns
#define Y00    0.28209479177387814f   // 1/(2*sqrt(pi)) = B[0,:,:]
#define PI_D   3.14159265358979323846
#define RSQRT_MUL 0.08838834764831845f   // 1/sqrt(128)

// output sections (float offsets), reference return order
#define OUT_COEFF 0                         // (16,4,64,36)   = 147456
#define OUT_POS   147456                    // (16,64,180,359)= 66170880
#define OUT_ANG   (147456 + 66170880)       // (16,1,180,359) = 1033920
#define OUT_RAD   (OUT_ANG + 1033920)       // (16,64)        = 1024

// workspace layout (float offsets)
#define WS_Y    0        // 180 GL nodes (padded 192)
#define WS_RL   192      // radial_logits copy (1024)
#define WS_AMAT 1216     // ang matrix 64x36 (2304)
#define WS_B    3520     // SH basis 36 x NPAD2

typedef float v2f __attribute__((ext_vector_type(2)));
typedef float v4f __attribute__((ext_vector_type(4)));
typedef float v8f __attribute__((ext_vector_type(8)));

// ---------------------------------------------------------------------------
// 1) Gauss-Legendre nodes for n=180 (Newton on P_180, double precision).
// ---------------------------------------------------------------------------
__global__ void k_glnodes(float* __restrict__ y) {
  const int n = RESB;
  int i = threadIdx.x;
  if (i < n / 2) {
    double z = cos(PI_D * ((double)i + 0.75) / ((double)n + 0.5));
    for (int it = 0; it < 64; ++it) {
      double p1 = 1.0, p2 = 0.0;
      for (int j = 1; j <= n; ++j) {
        double p3 = p2; p2 = p1;
        p1 = ((2.0 * j - 1.0) * z * p2 - (j - 1.0) * p3) / (double)j;
      }
      double pp = (double)n * (z * p1 - p2) / (z * z - 1.0);
      double zo = z;
      z = z - p1 / pp;
      if (fabs(z - zo) < 1e-15) break;
    }
    y[n - 1 - i] = (float)z;   // ascending order (numpy.leggauss)
    y[i]         = (float)(-z);
  }
}

// ---------------------------------------------------------------------------
// 2) Real SH basis B[k][ba] (k = l*l+l+m ordering), padded N, zero tail.
// ---------------------------------------------------------------------------
__device__ __forceinline__ double factd(int k) {
  double f = 1.0;
  for (int i = 2; i <= k; ++i) f *= (double)i;
  return f;
}

__global__ void k_basis(const float* __restrict__ y, float* __restrict__ B) {
  int ba = blockIdx.x * blockDim.x + threadIdx.x;
  if (ba >= NPAD2) return;
  if (ba >= NBA) {
    for (int k = 0; k < SHD; ++k) B[(size_t)k * NPAD2 + ba] = 0.0f;
    return;
  }
  int b = ba / RESA, a = ba % RESA;
  double yv = (double)y[b];
  double somx2 = sqrt(fmax(0.0, 1.0 - yv * yv));
  double P[6][6];
  P[0][0] = 1.0;
#pragma unroll
  for (int m = 1; m <= 5; ++m)
    P[m][m] = -(2.0 * m - 1.0) * somx2 * P[m - 1][m - 1];
#pragma unroll
  for (int m = 0; m <= 4; ++m)
    P[m + 1][m] = (2.0 * m + 1.0) * yv * P[m][m];
#pragma unroll
  for (int m = 0; m <= 5; ++m) {
#pragma unroll
    for (int l = m + 2; l <= 5; ++l)
      P[l][m] = ((2.0 * l - 1.0) * yv * P[l - 1][m] -
                 (double)(l + m - 1) * P[l - 2][m]) / (double)(l - m);
  }
  double alpha = 2.0 * PI_D * (double)a / (double)RESA;
#pragma unroll
  for (int l = 0; l <= 5; ++l) {
#pragma unroll
    for (int m = 0; m <= l; ++m) {
      double N = sqrt((2.0 * l + 1.0) / (4.0 * PI_D) * factd(l - m) / factd(l + m));
      if (m == 0) {
        B[(size_t)(l * l + l) * NPAD2 + ba] = (float)(N * P[l][0]);
      } else {
        double s = 1.4142135623730951 * N * P[l][m];
        B[(size_t)(l * l + l + m) * NPAD2 + ba] = (float)(s * cos(m * alpha));
        B[(size_t)(l * l + l - m) * NPAD2 + ba] = (float)(s * sin(m * alpha));
      }
    }
  }
}

// ---------------------------------------------------------------------------
// 3) Per-graph: gather focus node, species scaling, radial MLP, angular mix,
//    log_position_coeffs. One block (128 threads) per graph.
// ---------------------------------------------------------------------------
__global__ __launch_bounds__(128) void k_graph(
    const float* __restrict__ emb, const float* __restrict__ spec,
    const float* __restrict__ Wr, const float* __restrict__ W1,
    const float* __restrict__ W2,
    const float* __restrict__ Wa0, const float* __restrict__ Wa1,
    const float* __restrict__ Wa2, const float* __restrict__ Wa3,
    const float* __restrict__ Wa4, const float* __restrict__ Wa5,
    const int* __restrict__ focus, const int* __restrict__ tspec,
    float* __restrict__ out_coeff, float* __restrict__ out_rad,
    float* __restrict__ ws_rl, float* __restrict__ ws_amat) {
  __shared__ float s_e[EMBD];
  __shared__ float s_r[NRAD];
  __shared__ float s_h[MULK];
  __shared__ float s_ang[SHD * NCH];
  __shared__ float s_rl[NRAD];

  int g = blockIdx.x, t = threadIdx.x;
  const float* e  = emb  + (size_t)focus[g] * EMBD;
  const float* sp = spec + (size_t)tspec[g] * (MULK * 6);

  // scale each irrep by its species scalar
#pragma unroll
  for (int l = 0; l <= 5; ++l) {
    int d = 2 * l + 1, off = MULK * l * l;
    float s = sp[MULK * l + t];
    for (int j = 0; j < d; ++j)
      s_e[off + t * d + j] = e[off + t * d + j] * s;
  }
  __syncthreads();

  // r = (scal0 @ W_radial) / sqrt(128)
  if (t < NRAD) {
    float acc = 0.f;
    for (int k = 0; k < MULK; ++k) acc += s_e[k] * Wr[k * NRAD + t];
    s_r[t] = acc * RSQRT_MUL;
  }
  __syncthreads();

  // h = softplus((r @ W_mlp1)/8)
  {
    float acc = 0.f;
    for (int j = 0; j < NRAD; ++j) acc += s_r[j] * W1[j * MULK + t];
    acc *= 0.125f;
    s_h[t] = fmaxf(acc, 0.f) + log1pf(expf(-fabsf(acc)));
  }
  __syncthreads();

  // radial_logits = (h @ W_mlp2)/sqrt(128)
  if (t < NRAD) {
    float acc = 0.f;
    for (int m = 0; m < MULK; ++m) acc += s_h[m] * W2[m * NRAD + t];
    acc *= RSQRT_MUL;
    s_rl[t] = acc;
    ws_rl[g * NRAD + t]   = acc;
    out_rad[g * NRAD + t] = acc;
  }

  // ang[g][c][kk] = sum_k scaled[l][k][j] * W_ang_l[k][c] / sqrt(128)
  const float* Wa[6] = {Wa0, Wa1, Wa2, Wa3, Wa4, Wa5};
  for (int tt = t; tt < SHD * NCH; tt += 128) {
    int kk = tt >> 2, c = tt & 3;
    int l = (kk >= 25) ? 5 : (kk >= 16) ? 4 : (kk >= 9) ? 3 : (kk >= 4) ? 2 : (kk >= 1) ? 1 : 0;
    int j = kk - l * l, d = 2 * l + 1, off = MULK * l * l;
    const float* W = Wa[l];
    float acc = 0.f;
    for (int k = 0; k < MULK; ++k) acc += s_e[off + k * d + j] * W[k * NCH + c];
    acc *= RSQRT_MUL;
    s_ang[tt] = acc;                             // [kk*4 + c]
    ws_amat[(g * NCH + c) * SHD + kk] = acc;     // A matrix row = g*4+c
  }
  __syncthreads();

  // log_position_coeffs[g,c,r,kk] = ang + (kk==0)*radial_logits[r]
  for (int i = t; i < NCH * NRAD * SHD; i += 128) {
    int c = i / (NRAD * SHD), rem = i % (NRAD * SHD);
    int r = rem / SHD, kk = rem % SHD;
    float v = s_ang[kk * NCH + c] + (kk == 0 ? s_rl[r] : 0.f);
    out_coeff[(size_t)((g * NCH + c) * NRAD + r) * SHD + kk] = v;
  }
}

// ---------------------------------------------------------------------------
// 4) Fused WMMA SH-synthesis + channel logsumexp + radial broadcast stream.
//    Block = 8 waves = one (4-graph, 128-column) supertile.
//    Wave w: 16x16 WMMA tile -> in-register logsumexp over channels -> LDS.
//    Whole block then streams position_logits as coalesced b128 NT stores.
// ---------------------------------------------------------------------------
__global__ __launch_bounds__(256) void k_fused(
    const float* __restrict__ Amat, const float* __restrict__ B,
    const float* __restrict__ rl,
    float* __restrict__ out_pos, float* __restrict__ out_ang) {
  __shared__ float s_lse[NCH][128];   // lse per (graph-in-group, column)
  __shared__ float s_rad[NCH][NRAD];  // Y00 * radial_logits per (graph, r)

  int mt = blockIdx.x / NB_N;         // M-group: graphs 4*mt..4*mt+3
  int nb = blockIdx.x % NB_N;         // 128-column block
  int t  = threadIdx.x;
  int w  = t >> 5;                    // wave 0..7
  int lane = t & 31;
  int llo  = lane & 15;
  bool hi  = lane >= 16;
  int n0 = nb * 128 + w * 16;
  int m0 = mt * 16;

  // --- WMMA: 16x16 f32 tile, K=36 in 9 steps; EXEC all-ones throughout ---
  v8f acc = {};
#pragma unroll
  for (int kb = 0; kb < 9; ++kb) {
    int k0 = kb * 4 + (hi ? 2 : 0);
    v2f a, bb;
    a.x  = Amat[(m0 + llo) * SHD + k0];
    a.y  = Amat[(m0 + llo) * SHD + k0 + 1];
    bb.x = B[(size_t)k0 * NPAD2 + n0 + llo];
    bb.y = B[(size_t)(k0 + 1) * NPAD2 + n0 + llo];
    acc = __builtin_amdgcn_wmma_f32_16x16x4_f32(
        false, a, false, bb, (short)0, acc, false, false);
  }

  // rows v=0..3 -> graph (hi?2:0) c=0..3 ; rows v=4..7 -> graph +1
  float m1 = fmaxf(fmaxf(acc[0], acc[1]), fmaxf(acc[2], acc[3]));
  float l1 = m1 + logf(expf(acc[0] - m1) + expf(acc[1] - m1) +
                       expf(acc[2] - m1) + expf(acc[3] - m1));
  float m2 = fmaxf(fmaxf(acc[4], acc[5]), fmaxf(acc[6], acc[7]));
  float l2 = m2 + logf(expf(acc[4] - m2) + expf(acc[5] - m2) +
                       expf(acc[6] - m2) + expf(acc[7] - m2));
  int g4 = hi ? 2 : 0;
  s_lse[g4][w * 16 + llo]     = l1;
  s_lse[g4 + 1][w * 16 + llo] = l2;

  // stage Y00 * radial for the 4 graphs (exactly 256 values)
  {
    int gg = t >> 6, r = t & 63;
    s_rad[gg][r] = Y00 * rl[(mt * NCH + gg) * NRAD + r];
  }
  __syncthreads();

  int bab0 = nb * 128;

  // angular_logits: 4 x 128 values, NT scalar stores
  for (int i = t; i < NCH * 128; i += 256) {
    int gg = i >> 7, col = i & 127;
    int ba = bab0 + col;
    if (ba < NBA)
      __builtin_nontemporal_store(s_lse[gg][col],
                                  out_ang + (size_t)(mt * NCH + gg) * NBA + ba);
  }

  // position_logits: 4 g x 64 r x 128 ba, coalesced b128 NT stores.
  // One wave covers one full (g,r) row of 128 floats per iteration.
  for (int task = t; task < NCH * NRAD * 32; task += 256) {
    int q  = task & 31;
    int r  = (task >> 5) & 63;
    int gg = task >> 11;
    int ba4 = bab0 + q * 4;
    if (ba4 < NBA) {               // NBA % 4 == 0 -> clean float4 tail
      float radd = s_rad[gg][r];
      v4f lv = *(const v4f*)&s_lse[gg][q * 4];
      v4f v = {lv.x + radd, lv.y + radd, lv.z + radd, lv.w + radd};
      __builtin_nontemporal_store(
          v, (v4f*)(out_pos + (size_t)((mt * NCH + gg) * NRAD + r) * NBA + ba4));
    }
  }
}

// ---------------------------------------------------------------------------
extern "C" void kernel_launch(void* const* d_in, const int* in_sizes, int n_in,
                              void* d_out, int out_size, void* d_ws, size_t ws_size,
                              hipStream_t stream) {
  const float* emb  = (const float*)d_in[0];
  const float* spec = (const float*)d_in[1];
  const float* Wr   = (const float*)d_in[2];
  const float* W1   = (const float*)d_in[3];
  const float* W2   = (const float*)d_in[4];
  const float* Wa0  = (const float*)d_in[5];
  const float* Wa1  = (const float*)d_in[6];
  const float* Wa2  = (const float*)d_in[7];
  const float* Wa3  = (const float*)d_in[8];
  const float* Wa4  = (const float*)d_in[9];
  const float* Wa5  = (const float*)d_in[10];
  const int* focus  = (const int*)d_in[11];
  const int* tspec  = (const int*)d_in[12];

  float* out = (float*)d_out;
  float* ws  = (float*)d_ws;
  float* wsY  = ws + WS_Y;
  float* wsRL = ws + WS_RL;
  float* wsA  = ws + WS_AMAT;
  float* wsB  = ws + WS_B;

  k_glnodes<<<1, 128, 0, stream>>>(wsY);
  k_basis<<<(NPAD2 + 255) / 256, 256, 0, stream>>>(wsY, wsB);
  k_graph<<<NG, 128, 0, stream>>>(emb, spec, Wr, W1, W2,
                                  Wa0, Wa1, Wa2, Wa3, Wa4, Wa5,
                                  focus, tspec,
                                  out + OUT_COEFF, out + OUT_RAD, wsRL, wsA);
  k_fused<<<4 * NB_N, 256, 0, stream>>>(wsA, wsB, wsRL,
                                        out + OUT_POS, out + OUT_ANG);
}